// MultiNetwork_41695542509719
// MI455X (gfx1250) — compile-verified
//
#include <hip/hip_runtime.h>

typedef __attribute__((ext_vector_type(2))) float v2f;
typedef __attribute__((ext_vector_type(8))) float v8f;

#define NUM_NETWORKS 512
#define IN_F 63
#define HID 32
#define OUT_F 4
#define THREADS 256
#define WAVES 8

#define XSTRIDE 66   // x tile row stride (even -> aligned b64 A loads; 2m mod 64 distinct banks)
#define HSTRIDE 34   // h tile row stride (even; 34m mod 64 distinct banks)
#define SCRW    1088 // per-wave scratch floats: [0,1056) x-tile / [0,544) h0 ; [544,1088) h1

// fp32 WMMA: D(16x16) = A(16x4) * B(4x16) + C, full fp32 precision.
#define WMMA_F32(a, b, c) \
  __builtin_amdgcn_wmma_f32_16x16x4_f32(false, (a), false, (b), (short)0, (c), false, false)

#if __has_builtin(__builtin_amdgcn_global_load_async_to_lds_b32)
#define HAVE_ASYNC_LDS 1
typedef __attribute__((address_space(1))) int gint_t;   // global-AS int (builtin param type)
typedef __attribute__((address_space(3))) int lint_t;   // LDS-AS int
#else
#define HAVE_ASYNC_LDS 0
#endif

__device__ __forceinline__ void wait_asynccnt0() {
#if __has_builtin(__builtin_amdgcn_s_wait_asynccnt)
    __builtin_amdgcn_s_wait_asynccnt(0);
#else
    asm volatile("s_wait_asynccnt 0x0" ::: "memory");
#endif
}

__global__ __launch_bounds__(THREADS)
void mlp_grouped_kernel(const float* __restrict__ x,
                        const float* __restrict__ W0, const float* __restrict__ b0,
                        const float* __restrict__ W1, const float* __restrict__ b1,
                        const float* __restrict__ W2, const float* __restrict__ b2,
                        const int* __restrict__ counts,
                        float* __restrict__ out)
{
    // Weights in LDS, K-pair-interleaved: element (k,n) -> [(k>>1)*rowlen + 2n + (k&1)]
    // so each WMMA B fragment (W[k0][n], W[k0+1][n]) is one aligned ds_load_b64.
    __shared__ __align__(16) float sW0p[32 * 64];   // K padded 63->64 (zero row), 32 K-pairs x 32 n
    __shared__ float sb0[HID];
    __shared__ __align__(16) float sW1p[16 * 64];   // 16 K-pairs x 32 n
    __shared__ float sb1[HID];
    __shared__ __align__(16) float sW2p[16 * 32];   // 16 K-pairs x 16 n (N padded 4->16, zeros)
    __shared__ float sb2[OUT_F];
    __shared__ __align__(16) float sScr[WAVES][SCRW];
    __shared__ int sRed[THREADS];

    const int net = blockIdx.x;
    const int tid = threadIdx.x;

    // ---- cooperative load of this network's weights into LDS (pair-swizzled) ----
    for (int i = tid; i < 64 * HID; i += THREADS) {
        int k = i >> 5, n = i & 31;
        float v = (k < IN_F) ? W0[net * IN_F * HID + k * HID + n] : 0.0f;
        sW0p[(k >> 1) * 64 + n * 2 + (k & 1)] = v;
    }
    for (int i = tid; i < HID * HID; i += THREADS) {
        int k = i >> 5, n = i & 31;
        sW1p[(k >> 1) * 64 + n * 2 + (k & 1)] = W1[net * HID * HID + i];
    }
    for (int i = tid; i < HID * 16; i += THREADS) {
        int k = i >> 4, c = i & 15;
        float v = (c < OUT_F) ? W2[net * HID * OUT_F + k * OUT_F + c] : 0.0f;
        sW2p[(k >> 1) * 32 + c * 2 + (k & 1)] = v;
    }
    if (tid < HID) { sb0[tid] = b0[net * HID + tid]; sb1[tid] = b1[net * HID + tid]; }
    if (tid < OUT_F) sb2[tid] = b2[net * OUT_F + tid];

    // ---- start offset of this network: block-wide prefix sum over counts[0..net) ----
    int part = 0;
    for (int i = tid; i < net; i += THREADS) part += counts[i];
    sRed[tid] = part;
    __syncthreads();
    for (int s = THREADS / 2; s > 0; s >>= 1) {
        if (tid < s) sRed[tid] += sRed[tid + s];
        __syncthreads();
    }
    const int start = sRed[0];
    const int count = counts[net];
    const int numTiles = (count + 15) >> 4;

    const int wave = tid >> 5;
    const int lane = tid & 31;
    const int ln   = lane & 15;      // A row / B,C,D column for this lane
    const int hi   = lane >> 4;      // K / row-half selector per WMMA VGPR layout

    float* scr = &sScr[wave][0];     // x tile (stride 66) / h0 (stride 34) / out staging
    float* h1  = scr + 544;          // h1 (stride 34)

    for (int tt = wave; tt < numTiles; tt += WAVES) {
        const int row0 = tt << 4;
        const float* xg = x + (size_t)(start + row0) * IN_F;

#if HAVE_ASYNC_LDS
        // ---- async DMA global -> LDS for the 16x63 x-tile (ASYNCcnt path) ----
        asm volatile("s_wait_dscnt 0x0" ::: "memory");  // prior-iter LDS reads drained
        for (int i = lane; i < 16 * IN_F; i += 32) {
            int r = i / IN_F;
            int c = i - r * IN_F;
            if (row0 + r < count) {   // stale LDS in dead rows only affects unstored rows
                __builtin_amdgcn_global_load_async_to_lds_b32(
                    (gint_t*)(xg + i),
                    (lint_t*)(scr + r * XSTRIDE + c),
                    0, 0);
            }
        }
        if (lane < 16) scr[lane * XSTRIDE + 63] = 0.0f;   // zero K pad col
        wait_asynccnt0();
#else
        // ---- fallback: stage through VGPRs ----
        for (int i = lane; i < 16 * IN_F; i += 32) {
            int r = i / IN_F;
            int c = i - r * IN_F;
            float v = (row0 + r < count) ? xg[i] : 0.0f;
            scr[r * XSTRIDE + c] = v;
        }
        if (lane < 16) scr[lane * XSTRIDE + 63] = 0.0f;
#endif

        // Prefetch this wave's next x tile (global_prefetch_b8): 32 lanes x 128 B = 4 KB.
        {
            int nrow = row0 + 16 * WAVES;
            if (nrow + 15 < count) {
                const char* p = (const char*)(x + (size_t)(start + nrow) * IN_F) + lane * 128;
                __builtin_prefetch(p, 0, 1);
            }
        }
        __builtin_amdgcn_wave_barrier();   // same-wave LDS is in-order; fence code motion

        // ---- layer 0: [16x64(pad)] @ [64x32] -> two 16x16 f32 accumulators ----
        v8f acc0 = {}; v8f acc1 = {};
#pragma unroll
        for (int kk = 0; kk < 64; kk += 4) {
            const int k0 = kk + 2 * hi;              // even -> aligned b64
            const int p0 = (kk >> 1) + hi;           // K-pair index
            v2f a  = *(const v2f*)&scr[ln * XSTRIDE + k0];
            v2f bL = *(const v2f*)&sW0p[p0 * 64 + ln * 2];
            v2f bH = *(const v2f*)&sW0p[p0 * 64 + 32 + ln * 2];
            acc0 = WMMA_F32(a, bL, acc0);
            acc1 = WMMA_F32(a, bH, acc1);
        }
        __builtin_amdgcn_wave_barrier();

        // bias + ReLU -> h0 in LDS (stride 34); x tile is dead, reuse region
        {
            float bb0 = sb0[ln], bb1 = sb0[16 + ln];
#pragma unroll
            for (int e = 0; e < 8; ++e) {
                int r = e + 8 * hi;
                float v0 = acc0[e] + bb0; v0 = v0 > 0.0f ? v0 : 0.0f;
                float v1 = acc1[e] + bb1; v1 = v1 > 0.0f ? v1 : 0.0f;
                scr[r * HSTRIDE + ln] = v0;
                scr[r * HSTRIDE + 16 + ln] = v1;
            }
        }
        __builtin_amdgcn_wave_barrier();

        // ---- layer 1: [16x32] @ [32x32] ----
        v8f acc2 = {}; v8f acc3 = {};
#pragma unroll
        for (int kk = 0; kk < 32; kk += 4) {
            const int k0 = kk + 2 * hi;
            const int p0 = (kk >> 1) + hi;
            v2f a  = *(const v2f*)&scr[ln * HSTRIDE + k0];
            v2f bL = *(const v2f*)&sW1p[p0 * 64 + ln * 2];
            v2f bH = *(const v2f*)&sW1p[p0 * 64 + 32 + ln * 2];
            acc2 = WMMA_F32(a, bL, acc2);
            acc3 = WMMA_F32(a, bH, acc3);
        }
        __builtin_amdgcn_wave_barrier();

        // bias + ReLU -> h1 in LDS
        {
            float bb0 = sb1[ln], bb1 = sb1[16 + ln];
#pragma unroll
            for (int e = 0; e < 8; ++e) {
                int r = e + 8 * hi;
                float v0 = acc2[e] + bb0; v0 = v0 > 0.0f ? v0 : 0.0f;
                float v1 = acc3[e] + bb1; v1 = v1 > 0.0f ? v1 : 0.0f;
                h1[r * HSTRIDE + ln] = v0;
                h1[r * HSTRIDE + 16 + ln] = v1;
            }
        }
        __builtin_amdgcn_wave_barrier();

        // ---- layer 2: [16x32] @ [32x16(pad)] -> one accumulator (cols 0..3 valid) ----
        v8f acc4 = {};
#pragma unroll
        for (int kk = 0; kk < 32; kk += 4) {
            const int k0 = kk + 2 * hi;
            const int p0 = (kk >> 1) + hi;
            v2f a  = *(const v2f*)&h1[ln * HSTRIDE + k0];
            v2f bb = *(const v2f*)&sW2p[p0 * 32 + ln * 2];
            acc4 = WMMA_F32(a, bb, acc4);
        }
        __builtin_amdgcn_wave_barrier();

        // bias + stage 16x4 output tile in LDS
        {
            float ob = (ln < OUT_F) ? sb2[ln] : 0.0f;
#pragma unroll
            for (int e = 0; e < 8; ++e) {
                if (ln < OUT_F) scr[(e + 8 * hi) * OUT_F + ln] = acc4[e] + ob;
            }
        }
        __builtin_amdgcn_wave_barrier();

        // coalesced float4 stores: one 16 B row per lane
        if (lane < 16) {
            int grow = row0 + lane;
            if (grow < count) {
                float4 v = *(const float4*)(scr + lane * 4);
                *reinterpret_cast<float4*>(out + (size_t)(start + grow) * 4) = v;
            }
        }
        __builtin_amdgcn_wave_barrier();
    }
}

extern "C" void kernel_launch(void* const* d_in, const int* in_sizes, int n_in,
                              void* d_out, int out_size, void* d_ws, size_t ws_size,
                              hipStream_t stream) {
    const float* x  = (const float*)d_in[0];
    const float* W0 = (const float*)d_in[1];
    const float* b0 = (const float*)d_in[2];
    const float* W1 = (const float*)d_in[3];
    const float* b1 = (const float*)d_in[4];
    const float* W2 = (const float*)d_in[5];
    const float* b2 = (const float*)d_in[6];
    const int* counts = (const int*)d_in[7];
    float* out = (float*)d_out;

    mlp_grouped_kernel<<<NUM_NETWORKS, THREADS, 0, stream>>>(
        x, W0, b0, W1, b1, W2, b2, counts, out);

    (void)in_sizes; (void)n_in; (void)out_size; (void)d_ws; (void)ws_size;
}